// MLPPolicy_81071802679486
// MI455X (gfx1250) — compile-verified
//
#include <hip/hip_runtime.h>

#define B_SAMPLES 262144
#define T_ACTIONS 1048576

typedef __attribute__((ext_vector_type(2))) float v2f;
typedef __attribute__((ext_vector_type(8))) float v8f;

__device__ __forceinline__ v8f vzero8() {
  v8f z = {0.f, 0.f, 0.f, 0.f, 0.f, 0.f, 0.f, 0.f};
  return z;
}

// One wave computes a (NGROUPS*16) x (NTILES*16) tile of C = A @ B using
// V_WMMA_F32_16X16X4_F32.  A from LDS (row-major, stride lda), B from global
// (row-major, stride ldb, rows >= kmaxB read as zero).  Each B fragment is
// reused for NGROUPS A fragments, cutting L2 weight traffic by NGROUPS x.
template <int NTILES, int NGROUPS>
__device__ __forceinline__ void gemm16g(const float* A, int lda, int K,
                                        const float* __restrict__ Bm, int ldb,
                                        int kmaxB, v8f (&acc)[NGROUPS][NTILES]) {
  const int lane = threadIdx.x & 31;
  const int half = lane >> 4;  // 0: K={k0,k0+1}, 1: K={k0+2,k0+3}
  const int l = lane & 15;     // A: row-in-group, B: column
#pragma unroll
  for (int g = 0; g < NGROUPS; ++g)
#pragma unroll
    for (int t = 0; t < NTILES; ++t) acc[g][t] = vzero8();
  for (int k0 = 0; k0 < K; k0 += 4) {
    const int ka = k0 + 2 * half;
    v2f a[NGROUPS];
#pragma unroll
    for (int g = 0; g < NGROUPS; ++g) {
      a[g].x = A[(g * 16 + l) * lda + ka];
      a[g].y = A[(g * 16 + l) * lda + ka + 1];
    }
#pragma unroll
    for (int t = 0; t < NTILES; ++t) {
      const int n = t * 16 + l;
      v2f b;
      b.x = (ka < kmaxB) ? Bm[ka * ldb + n] : 0.f;
      b.y = (ka + 1 < kmaxB) ? Bm[(ka + 1) * ldb + n] : 0.f;
#pragma unroll
      for (int g = 0; g < NGROUPS; ++g)
        acc[g][t] = __builtin_amdgcn_wmma_f32_16x16x4_f32(
            false, a[g], false, b, (short)0, acc[g][t], false, false);
    }
  }
}

// Epilogue: C += bias (per output column), optional relu, store to a
// row-major [NGROUPS*16 x ldo] panel starting at column col0.
template <int NTILES, int NGROUPS>
__device__ __forceinline__ void store_bias_relu(
    float* out, int ldo, int col0, const float* __restrict__ bias,
    const v8f (&acc)[NGROUPS][NTILES], bool relu) {
  const int lane = threadIdx.x & 31;
  const int half = lane >> 4;
  const int l = lane & 15;
#pragma unroll
  for (int g = 0; g < NGROUPS; ++g)
#pragma unroll
    for (int t = 0; t < NTILES; ++t) {
      const float bv = bias[t * 16 + l];
#pragma unroll
      for (int r = 0; r < 8; ++r) {
        float v = acc[g][t][r] + bv;
        if (relu) v = fmaxf(v, 0.f);
        out[(g * 16 + r + 8 * half) * ldo + (col0 + t * 16 + l)] = v;
      }
    }
}

// ---------------------------------------------------------------------------
// Kernel 0: fold emb-LN + W_card into a gather table.
//   P[slot][id][n] = (LN(emb[id]) * ln1_g + ln1_b) @ W_card[slot*32:(slot+1)*32, n]
// 13*53*64 floats = 176 KB, permanently L2-resident afterwards.
// ---------------------------------------------------------------------------
__global__ __launch_bounds__(64) void precompute_card_kernel(
    const float* __restrict__ emb_table, const float* __restrict__ ln1_g,
    const float* __restrict__ ln1_b, const float* __restrict__ W_card,
    float* __restrict__ P) {
  const int slot = blockIdx.x / 53;
  const int id = blockIdx.x % 53;
  const int n = threadIdx.x;  // 0..63
  const float* er = emb_table + id * 32;
  float m = 0.f;
#pragma unroll
  for (int j = 0; j < 32; ++j) m += er[j];
  m *= (1.f / 32.f);
  float var = 0.f;
#pragma unroll
  for (int j = 0; j < 32; ++j) {
    const float d = er[j] - m;
    var += d * d;
  }
  var *= (1.f / 32.f);
  const float inv = rsqrtf(var + 1e-5f);
  float s = 0.f;
#pragma unroll
  for (int j = 0; j < 32; ++j) {
    const float ln = (er[j] - m) * inv * ln1_g[j] + ln1_b[j];
    s += ln * W_card[(slot * 32 + j) * 64 + n];
  }
  P[blockIdx.x * 64 + n] = s;
}

// ---------------------------------------------------------------------------
// Kernel 1: state encoder + value head.  One wave handles 32 samples
// (2 row-groups), so every weight fragment is reused twice.
// ---------------------------------------------------------------------------
__global__ __launch_bounds__(32) void state_encoder_kernel(
    const int* __restrict__ hand_ids, const float* __restrict__ dense_state,
    const float* __restrict__ P, const float* __restrict__ b_card,
    const float* __restrict__ W_lp, const float* __restrict__ b_lp,
    const float* __restrict__ W_sn, const float* __restrict__ b_sn,
    const float* __restrict__ W_ct, const float* __restrict__ b_ct,
    const float* __restrict__ W_sp, const float* __restrict__ b_sp,
    const float* __restrict__ ln2_g, const float* __restrict__ ln2_b,
    const float* __restrict__ Wv1, const float* __restrict__ bv1,
    const float* __restrict__ Wv2, const float* __restrict__ bv2,
    float* __restrict__ state_h_out, float* __restrict__ values_out) {
  __shared__ int s_ids[32 * 13];
  __shared__ __align__(16) float s_xs[32 * 108];  // dense rows, col 107 = 0
  __shared__ __align__(16) float s_x[32 * 224];   // concat(e, lp, sn, ct)
  __shared__ __align__(16) float s_h[32 * 64];    // hidden / normalized state
  __shared__ __align__(16) float s_hv[32 * 64];   // value-head hidden

  const int lane = threadIdx.x;
  const int base = blockIdx.x * 32;

  // stage hand ids (with PAD_ID remap) and dense rows
  for (int i = lane; i < 32 * 13; i += 32) {
    int id = hand_ids[base * 13 + i];
    s_ids[i] = (id < 0) ? 52 : id;
  }
  for (int i = lane; i < 32 * 107; i += 32) {
    const int s = i / 107, c = i % 107;
    s_xs[s * 108 + c] = dense_state[(base + s) * 107 + c];
  }
  s_xs[lane * 108 + 107] = 0.f;
  __syncthreads();

  // e = relu(b_card + sum_slot P[slot][id]) -> x[:, 0:64]  (float4 gathers)
  for (int idx = lane; idx < 32 * 16; idx += 32) {
    const int r = idx >> 4;
    const int c4 = (idx & 15) * 4;
    float4 a = *(const float4*)(b_card + c4);
#pragma unroll
    for (int slot = 0; slot < 13; ++slot) {
      const int id = s_ids[r * 13 + slot];
      const float4 p = *(const float4*)(P + (slot * 53 + id) * 64 + c4);
      a.x += p.x; a.y += p.y; a.z += p.z; a.w += p.w;
    }
    float4 o;
    o.x = fmaxf(a.x, 0.f); o.y = fmaxf(a.y, 0.f);
    o.z = fmaxf(a.z, 0.f); o.w = fmaxf(a.w, 0.f);
    *(float4*)(s_x + r * 224 + c4) = o;
  }

  v8f acc[2][4];
  v8f acc2[2][2];
  // lp = relu(dense[:, 0:52] @ W_lp)    -> x[:, 64:128]
  gemm16g<4, 2>(s_xs, 108, 52, W_lp, 64, 52, acc);
  store_bias_relu<4, 2>(s_x, 224, 64, b_lp, acc, true);
  // sn = relu(dense[:, 52:104] @ W_sn)  -> x[:, 128:192]
  gemm16g<4, 2>(s_xs + 52, 108, 52, W_sn, 64, 52, acc);
  store_bias_relu<4, 2>(s_x, 224, 128, b_sn, acc, true);
  // ct = relu(dense[:, 104:107] @ W_ct) -> x[:, 192:224]  (K padded 3->4)
  gemm16g<2, 2>(s_xs + 104, 108, 4, W_ct, 32, 3, acc2);
  store_bias_relu<2, 2>(s_x, 224, 192, b_ct, acc2, true);
  __syncthreads();

  // h = relu(x @ W_sp + b_sp)
  gemm16g<4, 2>(s_x, 224, 224, W_sp, 64, 224, acc);
  store_bias_relu<4, 2>(s_h, 64, 0, b_sp, acc, true);
  __syncthreads();

  // LayerNorm(64): each lane owns one of the 32 rows
  {
    float* row = s_h + lane * 64;
    float m = 0.f;
#pragma unroll
    for (int j = 0; j < 64; ++j) m += row[j];
    m *= (1.f / 64.f);
    float var = 0.f;
#pragma unroll
    for (int j = 0; j < 64; ++j) {
      const float d = row[j] - m;
      var += d * d;
    }
    var *= (1.f / 64.f);
    const float inv = rsqrtf(var + 1e-5f);
#pragma unroll
    for (int j = 0; j < 64; ++j)
      row[j] = (row[j] - m) * inv * ln2_g[j] + ln2_b[j];
  }
  __syncthreads();

  // publish state_h (float4)
  for (int i = lane; i < 32 * 16; i += 32)
    *((float4*)state_h_out + base * 16 + i) = ((const float4*)s_h)[i];

  // value head: relu(state_h @ Wv1 + bv1) @ Wv2 + bv2
  gemm16g<4, 2>(s_h, 64, 64, Wv1, 64, 64, acc);
  store_bias_relu<4, 2>(s_hv, 64, 0, bv1, acc, true);
  __syncthreads();
  {
    float s = bv2[0];
#pragma unroll
    for (int j = 0; j < 64; ++j) s += s_hv[lane * 64 + j] * Wv2[j];
    values_out[base + lane] = s;
  }
}

// ---------------------------------------------------------------------------
// Kernel 2: action encoder + policy head.  One wave handles 64 actions
// (4 row-groups), so every weight fragment is reused four times.
// ---------------------------------------------------------------------------
__global__ __launch_bounds__(32) void action_policy_kernel(
    const float* __restrict__ action_feats, const int* __restrict__ segment_ids,
    const float* __restrict__ Wa1, const float* __restrict__ ba1,
    const float* __restrict__ Wa2, const float* __restrict__ ba2,
    const float* __restrict__ Wp1, const float* __restrict__ bp1,
    const float* __restrict__ Wp2, const float* __restrict__ bp2,
    const float* __restrict__ state_h, float* __restrict__ logits) {
  __shared__ __align__(16) float s_joint[64 * 128];  // [state_h | aenc]
  __shared__ __align__(16) float s_a[64 * 64];       // hidden1, then policy hidden
  __shared__ __align__(16) float s_af[64 * 56];      // action feats, padded 53->56

  const int lane = threadIdx.x;
  const int base = blockIdx.x * 64;

  for (int i = lane; i < 64 * 56; i += 32) {
    const int s = i / 56, c = i % 56;
    s_af[i] = (c < 53) ? action_feats[(base + s) * 53 + c] : 0.f;
  }
  // gather each action's state row into joint[:, 0:64] (float4)
  for (int idx = lane; idx < 64 * 16; idx += 32) {
    const int r = idx >> 4;
    const int c4 = (idx & 15) * 4;
    const int seg = segment_ids[base + r];
    *(float4*)(s_joint + r * 128 + c4) = *(const float4*)(state_h + seg * 64 + c4);
  }
  __syncthreads();

  v8f acc[4][4];
  // a = relu(af @ Wa1 + ba1)   (K padded 53->56; weight rows >= 53 read 0)
  gemm16g<4, 4>(s_af, 56, 56, Wa1, 64, 53, acc);
  store_bias_relu<4, 4>(s_a, 64, 0, ba1, acc, true);
  __syncthreads();
  // aenc = relu(a @ Wa2 + ba2) -> joint[:, 64:128]
  gemm16g<4, 4>(s_a, 64, 64, Wa2, 64, 64, acc);
  store_bias_relu<4, 4>(s_joint, 128, 64, ba2, acc, true);
  __syncthreads();
  // p = relu(joint @ Wp1 + bp1)  (reuse s_a: hidden1 is dead now)
  gemm16g<4, 4>(s_joint, 128, 128, Wp1, 64, 128, acc);
  store_bias_relu<4, 4>(s_a, 64, 0, bp1, acc, true);
  __syncthreads();
  // logits = p @ Wp2 + bp2 : two rows per lane
  for (int r = lane; r < 64; r += 32) {
    float s = bp2[0];
#pragma unroll
    for (int j = 0; j < 64; ++j) s += s_a[r * 64 + j] * Wp2[j];
    logits[base + r] = s;
  }
}

extern "C" void kernel_launch(void* const* d_in, const int* in_sizes, int n_in,
                              void* d_out, int out_size, void* d_ws,
                              size_t ws_size, hipStream_t stream) {
  const int* hand_ids = (const int*)d_in[0];
  const float* dense = (const float*)d_in[1];
  const float* afeat = (const float*)d_in[2];
  const int* seg = (const int*)d_in[3];
  const float* emb = (const float*)d_in[4];
  const float* ln1g = (const float*)d_in[5];
  const float* ln1b = (const float*)d_in[6];
  const float* Wcard = (const float*)d_in[7];
  const float* bcard = (const float*)d_in[8];
  const float* Wlp = (const float*)d_in[9];
  const float* blp = (const float*)d_in[10];
  const float* Wsn = (const float*)d_in[11];
  const float* bsn = (const float*)d_in[12];
  const float* Wct = (const float*)d_in[13];
  const float* bct = (const float*)d_in[14];
  const float* Wsp = (const float*)d_in[15];
  const float* bsp = (const float*)d_in[16];
  const float* ln2g = (const float*)d_in[17];
  const float* ln2b = (const float*)d_in[18];
  const float* Wa1 = (const float*)d_in[19];
  const float* ba1 = (const float*)d_in[20];
  const float* Wa2 = (const float*)d_in[21];
  const float* ba2 = (const float*)d_in[22];
  const float* Wp1 = (const float*)d_in[23];
  const float* bp1 = (const float*)d_in[24];
  const float* Wp2 = (const float*)d_in[25];
  const float* bp2 = (const float*)d_in[26];
  const float* Wv1 = (const float*)d_in[27];
  const float* bv1 = (const float*)d_in[28];
  const float* Wv2 = (const float*)d_in[29];
  const float* bv2 = (const float*)d_in[30];

  float* logits = (float*)d_out;              // [T]
  float* values = (float*)d_out + T_ACTIONS;  // [B]
  float* state_h = (float*)d_ws;              // [B,64] f32 scratch (64 MB)
  float* P = state_h + (size_t)B_SAMPLES * 64;  // [13,53,64] card table (176 KB)

  precompute_card_kernel<<<13 * 53, 64, 0, stream>>>(emb, ln1g, ln1b, Wcard, P);

  state_encoder_kernel<<<B_SAMPLES / 32, 32, 0, stream>>>(
      hand_ids, dense, P, bcard, Wlp, blp, Wsn, bsn, Wct, bct, Wsp, bsp, ln2g,
      ln2b, Wv1, bv1, Wv2, bv2, state_h, values);

  action_policy_kernel<<<T_ACTIONS / 64, 32, 0, stream>>>(
      afeat, seg, Wa1, ba1, Wa2, ba2, Wp1, bp1, Wp2, bp2, state_h, logits);
}